// RGCN_DistMult_5574867550775
// MI455X (gfx1250) — compile-verified
//
#include <hip/hip_runtime.h>
#include <math.h>
#include <stdint.h>

// ---------------------------------------------------------------------------
// Problem constants (from reference)
// ---------------------------------------------------------------------------
#define NN   8192        // num nodes
#define HH   128         // hidden dim
#define EE   131072      // num edges
#define NBK  8           // num bases (block-diag)
#define SIK  16          // block size

typedef __attribute__((ext_vector_type(2))) float v2f;
typedef __attribute__((ext_vector_type(8))) float v8f;

// ---------------------------------------------------------------------------
// Blocked WMMA f32 GEMM (A row-major):  C[M x Nc] = epi(A @ B)
//   Block = 4 waves = 128 threads. Block tile = 256 rows x 16 cols.
//   Wave w computes rows [w*64, w*64+64) as 4 WMMA m-subtiles (one B fragment
//   feeds 4 v_wmma_f32_16x16x4_f32 issues).
//   B panel (32 x 16 f32 per chunk) is staged to LDS with
//   global_load_async_to_lds_b128 (ASYNCcnt-tracked), DOUBLE-BUFFERED so the
//   next chunk's staging overlaps the current chunk's WMMAs; fragments are
//   read back via ds_load.
//   Requires: M % 256 == 0, Nc % 16 == 0, K % 32 == 0, lda == K, ldc == Nc.
// ---------------------------------------------------------------------------
__global__ __launch_bounds__(128) void wmma_gemm_nn_blocked(
    const float* __restrict__ A,        // M x K, row-major
    const float* __restrict__ B,        // K x Nc, row-major (ldb = Nc)
    const float* __restrict__ bias,     // [Nc] or nullptr
    const float* __restrict__ addmat,   // [M x Nc] or nullptr
    float* __restrict__ C,
    int M, int Nc, int K, int relu)
{
    __shared__ float sB[2][32 * 16];    // double-buffered 32-K x 16-N panel

    const int lane = threadIdx.x;       // 0..31
    const int wave = threadIdx.y;       // 0..3
    const int t    = wave * 32 + lane;  // 0..127

    const int tilesAcross = Nc >> 4;
    const int blkM = (int)blockIdx.x / tilesAcross;   // 256-row block index
    const int blkN = (int)blockIdx.x % tilesAcross;   // 16-col block index
    const int colBase = blkN * 16;
    const int rowBase = blkM * 256 + wave * 64;

    const int half = lane >> 4;         // 0 or 1 (K pairing per ISA layout)
    const int l    = lane & 15;

    // staging assignment: thread t loads B[k0 + (t>>2)][colBase + (t&3)*4 ..+3]
    const int srow = t >> 2;            // 0..31
    const int scol = (t & 3) * 4;       // 0,4,8,12
    const uint32_t ldsOff  = (uint32_t)(srow * 16 + scol) * 4u;
    const uint32_t ldsDst0 = (uint32_t)(uintptr_t)&sB[0][0] + ldsOff;
    const uint32_t ldsDst1 = (uint32_t)(uintptr_t)&sB[1][0] + ldsOff;
    const float* bsrc = B + (long long)srow * Nc + colBase + scol;

    v8f c0 = {0,0,0,0,0,0,0,0};
    v8f c1 = {0,0,0,0,0,0,0,0};
    v8f c2 = {0,0,0,0,0,0,0,0};
    v8f c3 = {0,0,0,0,0,0,0,0};

    const float* __restrict__ a0 = A + (long long)(rowBase +  0 + l) * K;
    const float* __restrict__ a1 = A + (long long)(rowBase + 16 + l) * K;
    const float* __restrict__ a2 = A + (long long)(rowBase + 32 + l) * K;
    const float* __restrict__ a3 = A + (long long)(rowBase + 48 + l) * K;

    // prologue: stage chunk 0 into buffer 0
    {
        const uint64_t g = (uint64_t)(uintptr_t)bsrc;
        asm volatile("global_load_async_to_lds_b128 %0, %1, off"
                     :: "v"(ldsDst0), "v"(g) : "memory");
    }

    for (int k0 = 0; k0 < K; k0 += 32) {
        const int cur = (k0 >> 5) & 1;
        const float* __restrict__ sBc = &sB[cur][0];

        if (k0 + 32 < K) {
            // stage next chunk into the other buffer, then wait only for the
            // current chunk (async loads retire in order within a wave)
            const uint64_t g = (uint64_t)(uintptr_t)(bsrc + (long long)(k0 + 32) * Nc);
            const uint32_t d = cur ? ldsDst0 : ldsDst1;
            asm volatile("global_load_async_to_lds_b128 %0, %1, off"
                         :: "v"(d), "v"(g) : "memory");
            asm volatile("s_wait_asynccnt 1" ::: "memory");
        } else {
            asm volatile("s_wait_asynccnt 0" ::: "memory");
        }
        __syncthreads();   // all waves' staging for 'cur' now visible

        // ---- 8 K-steps of 4; one B fragment feeds 4 m-subtiles ----
#pragma unroll
        for (int kk = 0; kk < 32; kk += 4) {
            const int ka = kk + 2 * half;   // lanes 0-15: kk,kk+1 ; 16-31: kk+2,kk+3
            v2f b;
            b[0] = sBc[(ka + 0) * 16 + l];
            b[1] = sBc[(ka + 1) * 16 + l];

            // batch all 4 A-fragment loads (distinct regs -> clause + 1 wait)
            const float2 av0 = *(const float2*)(a0 + k0 + ka);
            const float2 av1 = *(const float2*)(a1 + k0 + ka);
            const float2 av2 = *(const float2*)(a2 + k0 + ka);
            const float2 av3 = *(const float2*)(a3 + k0 + ka);

            v2f a;
            a[0] = av0.x; a[1] = av0.y;
            c0 = __builtin_amdgcn_wmma_f32_16x16x4_f32(false, a, false, b, (short)0, c0, false, false);
            a[0] = av1.x; a[1] = av1.y;
            c1 = __builtin_amdgcn_wmma_f32_16x16x4_f32(false, a, false, b, (short)0, c1, false, false);
            a[0] = av2.x; a[1] = av2.y;
            c2 = __builtin_amdgcn_wmma_f32_16x16x4_f32(false, a, false, b, (short)0, c2, false, false);
            a[0] = av3.x; a[1] = av3.y;
            c3 = __builtin_amdgcn_wmma_f32_16x16x4_f32(false, a, false, b, (short)0, c3, false, false);
        }
        __syncthreads();   // protect the buffer that chunk k0+64 will overwrite
    }

    // ---- epilogue ----
    const int col = colBase + l;
    const float bv = bias ? bias[col] : 0.f;
#pragma unroll
    for (int i = 0; i < 4; ++i) {
        const v8f& c = (i == 0) ? c0 : (i == 1) ? c1 : (i == 2) ? c2 : c3;
#pragma unroll
        for (int r = 0; r < 8; ++r) {
            const int row = rowBase + i * 16 + r + 8 * half;
            float v = c[r] + bv;
            if (addmat) v += addmat[(long long)row * Nc + col];
            if (relu)   v = fmaxf(v, 0.f);
            C[(long long)row * Nc + col] = v;
        }
    }
}

// ---------------------------------------------------------------------------
// Simple WMMA f32 GEMM for the A^T @ B reductions (M = Nc = 128, K = 8192):
//   C[M x Nc] = A^T @ B with A stored K x M (A[m][k] = A[k*lda + m]).
//   One wave per 16x16 tile.
// ---------------------------------------------------------------------------
__global__ __launch_bounds__(128) void wmma_gemm_tn(
    const float* __restrict__ A, int lda,   // K x M storage
    const float* __restrict__ B, int ldb,   // K x Nc
    float* __restrict__ C,
    int M, int Nc, int K)
{
    const int lane = threadIdx.x;
    const int wave = threadIdx.y;
    const int tilesAcross = Nc >> 4;
    const int tile = blockIdx.x * 4 + wave;
    const int tm = tile / tilesAcross;
    const int tn = tile % tilesAcross;
    const int half = lane >> 4;
    const int l    = lane & 15;
    const int m = tm * 16 + l;
    const int n = tn * 16 + l;

    v8f c = {0,0,0,0,0,0,0,0};
    for (int k = 0; k < K; k += 4) {
        const int ka = k + 2 * half;
        v2f a, b;
        a[0] = A[(long long)(ka + 0) * lda + m];
        a[1] = A[(long long)(ka + 1) * lda + m];
        b[0] = B[(long long)(ka + 0) * ldb + n];
        b[1] = B[(long long)(ka + 1) * ldb + n];
        c = __builtin_amdgcn_wmma_f32_16x16x4_f32(false, a, false, b, (short)0, c, false, false);
    }
#pragma unroll
    for (int r = 0; r < 8; ++r) {
        const int row = tm * 16 + r + 8 * half;
        C[(long long)row * Nc + tn * 16 + l] = c[r];
    }
}

// ---------------------------------------------------------------------------
// Per-edge block-diagonal RGCN message + scatter-add:
//   msg[e] = (h[src[e]] as [8 x 16]) @ blockdiag(W[etype[e]]) * norm[e]
//   agg[dst[e]] += msg[e]
// One wave per edge; each lane computes 4 of the 128 output features.
// ---------------------------------------------------------------------------
__global__ __launch_bounds__(256) void edge_bdd_kernel(
    const float* __restrict__ h,
    const int* __restrict__ src,
    const int* __restrict__ dst,
    const int* __restrict__ et,
    const float* __restrict__ norm,
    const float* __restrict__ W,       // [NREL, 8, 16, 16]
    float* __restrict__ agg,
    int E)
{
    const int lane = threadIdx.x & 31;
    const int e = (int)((blockIdx.x * (long long)blockDim.x + threadIdx.x) >> 5);
    if (e >= E) return;

    const int   s   = src[e];
    const int   d   = dst[e];
    const int   r   = et[e];
    const float nrm = norm[e];

    const float* __restrict__ hs = h + (long long)s * HH;
    const float* __restrict__ Wr = W + (long long)r * (NBK * SIK * SIK);

#pragma unroll
    for (int j = 0; j < 4; ++j) {
        const int f = lane + 32 * j;       // output feature 0..127
        const int b = f >> 4;              // block index
        const int o = f & 15;              // col within block
        const float* __restrict__ hb = hs + b * 16;
        const float* __restrict__ wb = Wr + b * 256 + o;
        float acc = 0.f;
#pragma unroll
        for (int i = 0; i < 16; ++i)
            acc += hb[i] * wb[i * 16];
        atomicAdd(&agg[(long long)d * HH + f], acc * nrm);
    }
}

// ---------------------------------------------------------------------------
// Elementwise helpers
// ---------------------------------------------------------------------------
__global__ __launch_bounds__(256) void zero_f32_kernel(float* __restrict__ p, long long n)
{
    long long i = blockIdx.x * 256LL + threadIdx.x;
    if (i < n) p[i] = 0.f;
}

__global__ __launch_bounds__(256) void gather_rows_kernel(
    const float* __restrict__ emb, const int* __restrict__ ids,
    float* __restrict__ out, long long n)   // n = N*H
{
    long long i = blockIdx.x * 256LL + threadIdx.x;
    if (i < n) {
        int row = (int)(i / HH), col = (int)(i % HH);
        out[i] = emb[(long long)ids[row] * HH + col];
    }
}

// z = z_mean + exp(z_log_std) * eps   where zml = [z_mean | z_log_std] (N x 2H)
__global__ __launch_bounds__(256) void reparam_kernel(
    const float* __restrict__ zml, const float* __restrict__ eps,
    float* __restrict__ z, long long n)     // n = N*H
{
    long long i = blockIdx.x * 256LL + threadIdx.x;
    if (i < n) {
        int row = (int)(i / HH), col = (int)(i % HH);
        const float* zr = zml + (long long)row * (2 * HH);
        z[i] = zr[col] + expf(zr[HH + col]) * eps[i];
    }
}

// ---------------------------------------------------------------------------
// Launch orchestration
// ---------------------------------------------------------------------------
static inline void launch_gemm_nn(const float* A, const float* B,
                                  const float* bias, const float* addmat,
                                  float* C, int M, int Nc, int K, int relu,
                                  hipStream_t stream)
{
    dim3 blk(32, 4);
    dim3 grd((M / 256) * (Nc / 16));
    wmma_gemm_nn_blocked<<<grd, blk, 0, stream>>>(A, B, bias, addmat, C, M, Nc, K, relu);
}

static inline void launch_gemm_tn(const float* A, int lda, const float* B, int ldb,
                                  float* C, int M, int Nc, int K, hipStream_t stream)
{
    int tiles = (M / 16) * (Nc / 16);
    dim3 blk(32, 4);
    dim3 grd(tiles / 4);
    wmma_gemm_tn<<<grd, blk, 0, stream>>>(A, lda, B, ldb, C, M, Nc, K);
}

extern "C" void kernel_launch(void* const* d_in, const int* in_sizes, int n_in,
                              void* d_out, int out_size, void* d_ws, size_t ws_size,
                              hipStream_t stream)
{
    // inputs in setup_inputs() order
    const int*   node_ids = (const int*)  d_in[0];
    const int*   src      = (const int*)  d_in[1];
    const int*   dst      = (const int*)  d_in[2];
    const int*   etypes   = (const int*)  d_in[3];
    const float* norm     = (const float*)d_in[4];
    const float* eps      = (const float*)d_in[5];
    const float* emb      = (const float*)d_in[6];
    const float* W0       = (const float*)d_in[7];
    const float* loop0    = (const float*)d_in[8];
    const float* b0       = (const float*)d_in[9];
    const float* W1       = (const float*)d_in[10];
    const float* loop1    = (const float*)d_in[11];
    const float* b1       = (const float*)d_in[12];
    const float* Wz       = (const float*)d_in[13];
    const float* bz       = (const float*)d_in[14];
    const float* Wi       = (const float*)d_in[15];
    const float* bi       = (const float*)d_in[16];
    const float* hbi      = (const float*)d_in[17];
    const float* Wo       = (const float*)d_in[18];
    const float* bo       = (const float*)d_in[19];
    const float* hbo      = (const float*)d_in[20];

    float* out = (float*)d_out;

    const long long NH = (long long)NN * HH;
    float* ws  = (float*)d_ws;
    float* x0  = ws;                 // N*H
    float* h0  = x0  + NH;           // N*H
    float* h1  = h0  + NH;           // N*H
    float* agg = h1  + NH;           // N*H
    float* zml = agg + NH;           // N*2H
    float* z   = zml + 2 * NH;       // N*H
    float* v   = z   + NH;           // N*H
    float* t   = v   + NH;           // H*H
    float* h2  = t   + (long long)HH * HH;  // N*H

    const int TPB = 256;
    dim3 ew_grd((EE * 32 + TPB - 1) / TPB);   // one wave per edge
    dim3 el_grd((int)((NH + TPB - 1) / TPB));

    // ---- input embedding lookup:  x0 = embedding[node_ids] ----
    gather_rows_kernel<<<el_grd, TPB, 0, stream>>>(emb, node_ids, x0, NH);

    // ---- RGCN layer 0:  h0 = relu(agg0 + b0 + x0 @ loop0) ----
    zero_f32_kernel<<<el_grd, TPB, 0, stream>>>(agg, NH);
    edge_bdd_kernel<<<ew_grd, TPB, 0, stream>>>(x0, src, dst, etypes, norm, W0, agg, EE);
    launch_gemm_nn(x0, loop0, b0, agg, h0, NN, HH, HH, 1, stream);

    // ---- RGCN layer 1:  h1 = agg1 + b1 + h0 @ loop1 ----
    zero_f32_kernel<<<el_grd, TPB, 0, stream>>>(agg, NH);
    edge_bdd_kernel<<<ew_grd, TPB, 0, stream>>>(h0, src, dst, etypes, norm, W1, agg, EE);
    launch_gemm_nn(h0, loop1, b1, agg, h1, NN, HH, HH, 0, stream);

    // ---- VAE head:  zml = h1 @ Wz + bz ;  z = mean + exp(log_std)*eps ----
    launch_gemm_nn(h1, Wz, bz, nullptr, zml, NN, 2 * HH, HH, 0, stream);
    reparam_kernel<<<el_grd, TPB, 0, stream>>>(zml, eps, z, NH);

    // ---- decode 1 (algebraic fusion):
    //   R@(z@Wi + bi) + R@x0 + hbi  ==  z @ (z^T @ (z@Wi + bi + x0)) + hbi
    //   (avoids the 8192x8192 R matrix entirely: 256 MB > 192 MB L2)
    launch_gemm_nn(z, Wi, bi, x0, v, NN, HH, HH, 0, stream);          // v = z@Wi+bi+x0
    launch_gemm_tn(z, HH, v, HH, t, HH, HH, NN, stream);              // t = z^T @ v
    launch_gemm_nn(z, t, hbi, nullptr, h2, NN, HH, HH, 0, stream);    // h2 = z@t + hbi

    // ---- decode 2:  out = h2 @ (h2^T @ (h2@Wo + bo + x0)) + hbo ----
    launch_gemm_nn(h2, Wo, bo, x0, v, NN, HH, HH, 0, stream);
    launch_gemm_tn(h2, HH, v, HH, t, HH, HH, NN, stream);
    launch_gemm_nn(h2, t, hbo, nullptr, out, NN, HH, HH, 0, stream);

    (void)in_sizes; (void)n_in; (void)out_size; (void)ws_size;
}